// SelfAttention_9534827397170
// MI455X (gfx1250) — compile-verified
//
#include <hip/hip_runtime.h>
#include <hip/hip_bf16.h>

// Problem constants (reference: B=4, L=512, H=256, D=128)
#define B_ 4
#define L_ 512
#define H_ 256
#define D_ 128

typedef __attribute__((ext_vector_type(2))) float v2f;
typedef __attribute__((ext_vector_type(8))) float v8f;

// gfx1250 has a hardware transcendental tanh (V_TANH_F32).
__device__ __forceinline__ float tanh_hw(float x) {
#if __has_builtin(__builtin_amdgcn_tanhf)
    return __builtin_amdgcn_tanhf(x);
#else
    float r;
    asm volatile("v_tanh_f32 %0, %1" : "=v"(r) : "v"(x));
    return r;
#endif
}

__device__ __forceinline__ float exp2_hw(float x) {
#if __has_builtin(__builtin_amdgcn_exp2f)
    return __builtin_amdgcn_exp2f(x);
#else
    return exp2f(x);
#endif
}

// ---------------------------------------------------------------------------
// Kernel 1: dual projection GEMM with native-f32 WMMA.
//   wc[m,n] = sum_k c[m,k] * Wc[n,k]    (m = b*L+l flattened: 2048x256 @ 256x128)
//   ww[m,n] = sum_k c[m,k] * Ww[n,k]
// One wave computes one 16x16 tile of BOTH outputs (shared A fragment).
// Tiles: (2048/16) * (128/16) = 1024 waves -> 128 blocks x 8 waves.
// V_WMMA_F32_16X16X4_F32: A is 16x4 f32 (2 VGPRs/lane), B is 4x16 f32,
// C/D is 16x16 f32 (8 VGPRs/lane). K=256 -> 64 WMMA steps per output pair.
// ---------------------------------------------------------------------------
__global__ __launch_bounds__(256) void dual_gemm_wmma(
    const float* __restrict__ c,    // [2048, 256]
    const float* __restrict__ Wc,   // [128, 256]
    const float* __restrict__ Ww,   // [128, 256]
    float* __restrict__ wc_out,     // [2048, 128]
    float* __restrict__ ww_out)     // [2048, 128]
{
    const int lane = threadIdx.x & 31;
    const int wid  = threadIdx.x >> 5;
    const int tile = blockIdx.x * 8 + wid;        // 0..1023
    const int mt   = tile >> 3;                   // 0..127
    const int nt   = tile & 7;                    // 0..7
    const int m0   = mt * 16;
    const int n0   = nt * 16;
    const int row  = lane & 15;                   // A: row M, B: col N
    const int half = lane >> 4;                   // lanes 16..31 hold K+2

    v8f accC = {0.f, 0.f, 0.f, 0.f, 0.f, 0.f, 0.f, 0.f};
    v8f accW = {0.f, 0.f, 0.f, 0.f, 0.f, 0.f, 0.f, 0.f};

#if __has_builtin(__builtin_amdgcn_wmma_f32_16x16x4_f32)
    // Per ISA layout:
    //  A 16x4 f32 : lanes 0-15 -> M=lane, VGPR0/1 = K0/K1; lanes 16-31 -> K2/K3
    //  B 4x16 f32 : lanes 0-15 -> N=lane, VGPR0/1 = K0/K1; lanes 16-31 -> K2/K3
    const float* aPtr  = c  + (size_t)(m0 + row) * H_ + half * 2;
    const float* bcPtr = Wc + (size_t)(n0 + row) * H_ + half * 2;
    const float* bwPtr = Ww + (size_t)(n0 + row) * H_ + half * 2;
    #pragma unroll 4
    for (int k = 0; k < H_ / 4; ++k) {
        v2f a  = *(const v2f*)(aPtr  + k * 4);
        v2f bc = *(const v2f*)(bcPtr + k * 4);
        v2f bw = *(const v2f*)(bwPtr + k * 4);
        accC = __builtin_amdgcn_wmma_f32_16x16x4_f32(
            false, a, false, bc, (short)0, accC, false, false);
        accW = __builtin_amdgcn_wmma_f32_16x16x4_f32(
            false, a, false, bw, (short)0, accW, false, false);
    }
#else
    // Scalar fallback (keeps compile green if the f32 WMMA builtin is absent).
    for (int r = 0; r < 8; ++r) {
        const int m = m0 + r + half * 8;
        const int n = n0 + row;
        float sc = 0.f, sw = 0.f;
        for (int k = 0; k < H_; ++k) {
            const float cv = c[(size_t)m * H_ + k];
            sc += cv * Wc[(size_t)n * H_ + k];
            sw += cv * Ww[(size_t)n * H_ + k];
        }
        accC[r] = sc;
        accW[r] = sw;
    }
#endif

    // C/D layout: VGPR r, lanes 0-15 -> (M=r, N=lane); lanes 16-31 -> (M=8+r).
    #pragma unroll
    for (int r = 0; r < 8; ++r) {
        const int m = m0 + r + half * 8;
        const int n = n0 + row;
        wc_out[(size_t)m * D_ + n] = accC[r];
        ww_out[(size_t)m * D_ + n] = accW[r];
    }
}

// ---------------------------------------------------------------------------
// Kernel 2: fused scores(tanh) -> softmax -> h = a @ c, one block per (b,i).
//   s_j = sum_d v[d] * tanh(wc[b,j,d] + ww[b,i,d])    (hardware v_tanh_f32)
//   a_j = softmax_j(s)                                 (v_exp_f32 via exp2)
//   h[b,i,t] = sum_j a_j * c[b,j,t]                    (t = 0..255)
// 8 waves/block: wave w computes s_j for j = jt*8+w; lane holds d=4*lane..+3.
// ---------------------------------------------------------------------------
__global__ __launch_bounds__(256) void attn_fused(
    const float* __restrict__ c,    // [B, L, H]
    const float* __restrict__ Wv,   // [1, 128]
    const float* __restrict__ wc,   // [2048, 128]  (c @ Wc^T)
    const float* __restrict__ ww,   // [2048, 128]  (c @ Ww^T)
    float* __restrict__ out)        // [B, L, H]
{
    __shared__ float s[L_];
    __shared__ float red[256];

    const int tid  = threadIdx.x;
    const int lane = tid & 31;
    const int wid  = tid >> 5;
    const int bi   = blockIdx.x;          // b*512 + i
    const int b    = bi >> 9;

    // Lane-private slice: d = 4*lane .. 4*lane+3 (float4, 512B/wave coalesced)
    const float4 v4 = *(const float4*)(Wv + lane * 4);
    const float4 w4 = *(const float4*)(ww + (size_t)bi * D_ + lane * 4);
    const float* wcb = wc + (size_t)b * L_ * D_;

    const float LOG2E = 1.4426950408889634f;

    // ---- scores: 512 j's, 8 per block-iteration (one per wave) ----
    for (int jt = 0; jt < L_ / 8; ++jt) {
        const int j = jt * 8 + wid;
        const float4 x = *(const float4*)(wcb + (size_t)j * D_ + lane * 4);
        float dot = v4.x * tanh_hw(x.x + w4.x)
                  + v4.y * tanh_hw(x.y + w4.y)
                  + v4.z * tanh_hw(x.z + w4.z)
                  + v4.w * tanh_hw(x.w + w4.w);
        #pragma unroll
        for (int off = 16; off > 0; off >>= 1)
            dot += __shfl_xor(dot, off, 32);
        if (lane == 0) s[j] = dot;
    }
    __syncthreads();

    // ---- softmax over 512 (each thread owns 2 entries) ----
    const float a0 = s[tid];
    const float a1 = s[tid + 256];
    red[tid] = fmaxf(a0, a1);
    __syncthreads();
    for (int off = 128; off > 0; off >>= 1) {
        if (tid < off) red[tid] = fmaxf(red[tid], red[tid + off]);
        __syncthreads();
    }
    const float mx = red[0];
    __syncthreads();

    const float p0 = exp2_hw((a0 - mx) * LOG2E);
    const float p1 = exp2_hw((a1 - mx) * LOG2E);
    s[tid]       = p0;
    s[tid + 256] = p1;
    red[tid] = p0 + p1;
    __syncthreads();
    for (int off = 128; off > 0; off >>= 1) {
        if (tid < off) red[tid] += red[tid + off];
        __syncthreads();
    }
    const float inv = 1.0f / red[0];

    // ---- h[b,i,tid] = inv * sum_j p_j * c[b,j,tid]  (c is L2-resident) ----
    const float* cb = c + (size_t)b * L_ * H_ + tid;
    float acc0 = 0.f, acc1 = 0.f, acc2 = 0.f, acc3 = 0.f;
    for (int j = 0; j < L_; j += 4) {
        acc0 += s[j + 0] * cb[(size_t)(j + 0) * H_];
        acc1 += s[j + 1] * cb[(size_t)(j + 1) * H_];
        acc2 += s[j + 2] * cb[(size_t)(j + 2) * H_];
        acc3 += s[j + 3] * cb[(size_t)(j + 3) * H_];
    }
    out[(size_t)bi * H_ + tid] = inv * ((acc0 + acc1) + (acc2 + acc3));
}

// ---------------------------------------------------------------------------
// Host entry
// ---------------------------------------------------------------------------
extern "C" void kernel_launch(void* const* d_in, const int* in_sizes, int n_in,
                              void* d_out, int out_size, void* d_ws, size_t ws_size,
                              hipStream_t stream) {
    const float* c  = (const float*)d_in[0];   // [B,L,H] f32
    // d_in[1] = c_mask (bool) — accepted but unused by the reference forward
    const float* Wc = (const float*)d_in[2];   // [D,H] f32
    const float* Ww = (const float*)d_in[3];   // [D,H] f32
    const float* Wv = (const float*)d_in[4];   // [1,D] f32
    float* out = (float*)d_out;                // [B,L,H] f32

    float* wc_ws = (float*)d_ws;                         // 2048*128 f32 (1 MB)
    float* ww_ws = wc_ws + (size_t)B_ * L_ * D_;         // 2048*128 f32 (1 MB)

    // 1024 output tiles / 8 waves per block
    dual_gemm_wmma<<<128, 256, 0, stream>>>(c, Wc, Ww, wc_ws, ww_ws);
    // one block per (b, i)
    attn_fused<<<B_ * L_, 256, 0, stream>>>(c, Wv, wc_ws, ww_ws, out);
}